// GraphTransformer_35304631173352
// MI455X (gfx1250) — compile-verified
//
#include <hip/hip_runtime.h>
#include <hip/hip_bf16.h>

#define NN   50000
#define EE   800000
#define FIN_ 128
#define PE_  32
#define HID_ 128
#define OUT_ 64

typedef float v2f __attribute__((ext_vector_type(2)));
typedef float v8f __attribute__((ext_vector_type(8)));

__device__ __forceinline__ v8f wmma4(v2f a, v2f b, v8f c) {
    // D = A(16x4,f32) * B(4x16,f32) + C(16x16,f32)
    return __builtin_amdgcn_wmma_f32_16x16x4_f32(false, a, false, b, (short)0, c,
                                                 false, false);
}

// ---------------------------------------------------------------------------
// Generic fp32 WMMA GEMM: C[N x P] = A[N x K] @ W[K x P] (+bias) (+relu)
// One wave per 16x16 output tile. 8 waves per block stacked over row tiles.
// ---------------------------------------------------------------------------
template<int K, bool BIAS, bool RELU>
__global__ __launch_bounds__(256) void gemm_wmma_k(
    const float* __restrict__ A, const float* __restrict__ W,
    const float* __restrict__ bias, float* __restrict__ C,
    int nRowTiles, int P)
{
    const int wave = threadIdx.x >> 5;
    const int lane = threadIdx.x & 31;
    const int m    = lane & 15;
    const int kh   = lane >> 4;
    const int rt   = blockIdx.x * 8 + wave;
    if (rt >= nRowTiles) return;
    const int colBase = blockIdx.y << 4;
    const float* Arow = A + (size_t)(rt * 16 + m) * K;
    const int col = colBase + m;

    v8f acc = {};
#pragma unroll 8
    for (int k0 = 0; k0 < K; k0 += 4) {
        const int ka = k0 + 2 * kh;
        v2f a = *(const v2f*)(Arow + ka);
        v2f b;
        b.x = W[(ka    ) * P + col];
        b.y = W[(ka + 1) * P + col];
        acc = wmma4(a, b, acc);
    }
    float bb = 0.f;
    if (BIAS) bb = bias[col];
#pragma unroll
    for (int r = 0; r < 8; ++r) {
        float val = acc[r] + bb;
        if (RELU) val = val > 0.f ? val : 0.f;
        C[(size_t)(rt * 16 + r + 8 * kh) * P + col] = val;
    }
}

// ---------------------------------------------------------------------------
// Input projection: concat([x, pos], -1)[N x 160] @ W_in[160 x 128] + b_in
// ---------------------------------------------------------------------------
__global__ __launch_bounds__(256) void gemm_inproj(
    const float* __restrict__ X, const float* __restrict__ Ppos,
    const float* __restrict__ W, const float* __restrict__ bias,
    float* __restrict__ C, int nRowTiles)
{
    const int wave = threadIdx.x >> 5;
    const int lane = threadIdx.x & 31;
    const int m    = lane & 15;
    const int kh   = lane >> 4;
    const int rt   = blockIdx.x * 8 + wave;
    if (rt >= nRowTiles) return;
    const int col = (blockIdx.y << 4) + m;
    const int row = rt * 16 + m;

    v8f acc = {};
#pragma unroll 8
    for (int k0 = 0; k0 < 160; k0 += 4) {
        const int ka = k0 + 2 * kh;
        v2f a;
        if (ka < FIN_) a = *(const v2f*)(X    + (size_t)row * FIN_ + ka);
        else           a = *(const v2f*)(Ppos + (size_t)row * PE_  + (ka - FIN_));
        v2f b;
        b.x = W[(ka    ) * HID_ + col];
        b.y = W[(ka + 1) * HID_ + col];
        acc = wmma4(a, b, acc);
    }
    const float bb = bias[col];
#pragma unroll
    for (int r = 0; r < 8; ++r)
        C[(size_t)(rt * 16 + r + 8 * kh) * HID_ + col] = acc[r] + bb;
}

// ---------------------------------------------------------------------------
// Fused FFN: f = relu(h @ W1 + b1) @ W2 + b2, intermediate staged in LDS.
// 8 waves, 128 rows per block; LDS tile 128 x 516 floats (264192 B).
// ---------------------------------------------------------------------------
__global__ __launch_bounds__(256) void ffn_fused(
    const float* __restrict__ h,  const float* __restrict__ W1,
    const float* __restrict__ b1, const float* __restrict__ W2,
    const float* __restrict__ b2, float* __restrict__ f, int nRowTiles)
{
    extern __shared__ float t[];          // [128][516]
    const int LDT  = 516;                 // pad: stride % 64 banks == 4
    const int wave = threadIdx.x >> 5;
    const int lane = threadIdx.x & 31;
    const int m    = lane & 15;
    const int kh   = lane >> 4;
    const int rt   = blockIdx.x * 8 + wave;
    const bool valid = rt < nRowTiles;
    const int  rtc   = valid ? rt : (nRowTiles - 1);
    const float* Arow = h + (size_t)(rtc * 16 + m) * HID_;

    // GEMM1: t = relu(h @ W1 + b1), 16 rows x 512 cols per wave, K=128
    for (int ct = 0; ct < 32; ++ct) {
        const int col = ct * 16 + m;
        v8f acc = {};
#pragma unroll 8
        for (int k0 = 0; k0 < 128; k0 += 4) {
            const int ka = k0 + 2 * kh;
            v2f a = *(const v2f*)(Arow + ka);
            v2f b;
            b.x = W1[(ka    ) * 512 + col];
            b.y = W1[(ka + 1) * 512 + col];
            acc = wmma4(a, b, acc);
        }
        const float bb = b1[col];
#pragma unroll
        for (int r = 0; r < 8; ++r) {
            float val = acc[r] + bb;
            t[(wave * 16 + r + 8 * kh) * LDT + col] = val > 0.f ? val : 0.f;
        }
    }
    __syncthreads();

    // GEMM2: f = t @ W2 + b2, K=512 from LDS, 16 rows x 128 cols per wave
    const float* trow = t + (wave * 16 + m) * LDT;
    for (int ct = 0; ct < 8; ++ct) {
        const int col = ct * 16 + m;
        v8f acc = {};
#pragma unroll 8
        for (int k0 = 0; k0 < 512; k0 += 4) {
            const int ka = k0 + 2 * kh;
            v2f a = *(const v2f*)(trow + ka);          // ds_load_b64
            v2f b;
            b.x = W2[(ka    ) * 128 + col];
            b.y = W2[(ka + 1) * 128 + col];
            acc = wmma4(a, b, acc);
        }
        if (valid) {
            const float bb = b2[col];
#pragma unroll
            for (int r = 0; r < 8; ++r)
                f[(size_t)(rt * 16 + r + 8 * kh) * 128 + col] = acc[r] + bb;
        }
    }
}

// ---------------------------------------------------------------------------
// Edge attention: one wave per edge. Lane L owns feature 32*j + L of head j.
// ---------------------------------------------------------------------------
__global__ __launch_bounds__(256) void edge_attn(
    const int* __restrict__ rowIdx, const int* __restrict__ colIdx,
    const float* __restrict__ q, const float* __restrict__ k,
    const float* __restrict__ v, float* __restrict__ agg, int nE)
{
    const int e = blockIdx.x * 8 + (threadIdx.x >> 5);
    if (e >= nE) return;
    const int lane = threadIdx.x & 31;
    const int r = rowIdx[e];
    const int c = colIdx[e];
    const float* qr = q + (size_t)r * HID_;
    const float* kc = k + (size_t)c * HID_;
    const float* vc = v + (size_t)c * HID_;

    float dot[4];
#pragma unroll
    for (int j = 0; j < 4; ++j) {
        float p = qr[32 * j + lane] * kc[32 * j + lane];
#pragma unroll
        for (int off = 16; off > 0; off >>= 1) p += __shfl_xor(p, off, 32);
        dot[j] = p;
    }
    const float scale = 0.17677669529663687f;   // 32^-0.5
    float a0[4], mx = -3.402823466e38f;
#pragma unroll
    for (int j = 0; j < 4; ++j) {
        float a = dot[j] * scale;
        a = a >= 0.f ? a : 0.2f * a;            // leaky_relu
        a0[j] = a;
        mx = fmaxf(mx, a);
    }
    float s = 0.f;
#pragma unroll
    for (int j = 0; j < 4; ++j) { a0[j] = __expf(a0[j] - mx); s += a0[j]; }
    const float inv = 1.f / s;
#pragma unroll
    for (int j = 0; j < 4; ++j) {
        const float w = a0[j] * inv;
        unsafeAtomicAdd(&agg[(size_t)r * HID_ + 32 * j + lane],
                        w * vc[32 * j + lane]);
    }
}

// ---------------------------------------------------------------------------
// h = LayerNorm(h + d) * g + b  (one wave per row, 4 features per lane)
// ---------------------------------------------------------------------------
__global__ __launch_bounds__(256) void add_ln(
    float* __restrict__ h, const float* __restrict__ d,
    const float* __restrict__ g, const float* __restrict__ b, int n)
{
    const int row = blockIdx.x * 8 + (threadIdx.x >> 5);
    if (row >= n) return;
    const int lane = threadIdx.x & 31;
    const size_t base = (size_t)row * HID_ + lane * 4;
    float4 hv = *(const float4*)(h + base);
    float4 dv = *(const float4*)(d + base);
    float y0 = hv.x + dv.x, y1 = hv.y + dv.y, y2 = hv.z + dv.z, y3 = hv.w + dv.w;
    float s  = y0 + y1 + y2 + y3;
    float sq = y0 * y0 + y1 * y1 + y2 * y2 + y3 * y3;
#pragma unroll
    for (int off = 16; off > 0; off >>= 1) {
        s  += __shfl_xor(s,  off, 32);
        sq += __shfl_xor(sq, off, 32);
    }
    const float mean = s * (1.f / 128.f);
    const float var  = sq * (1.f / 128.f) - mean * mean;
    const float rn   = rsqrtf(var + 1e-5f);
    float4 gg = *(const float4*)(g + lane * 4);
    float4 bb = *(const float4*)(b + lane * 4);
    float4 o;
    o.x = (y0 - mean) * rn * gg.x + bb.x;
    o.y = (y1 - mean) * rn * gg.y + bb.y;
    o.z = (y2 - mean) * rn * gg.z + bb.z;
    o.w = (y3 - mean) * rn * gg.w + bb.w;
    *(float4*)(h + base) = o;
}

__global__ void zero_f32(float* __restrict__ p, long n) {
    long i = (long)blockIdx.x * blockDim.x + threadIdx.x;
    if (i < n) p[i] = 0.f;
}

// Weff[i][j] = sum_c W_f1[c*128 + i][j]   (tile(h,4) @ W_f1 == h @ Weff)
__global__ void sum_wf1(const float* __restrict__ Wf1, float* __restrict__ Weff) {
    int i = blockIdx.x * blockDim.x + threadIdx.x;
    if (i < 128 * 128) {
        int r = i >> 7, c = i & 127;
        float s = 0.f;
#pragma unroll
        for (int blk = 0; blk < 4; ++blk) s += Wf1[((blk << 7) + r) * 128 + c];
        Weff[i] = s;
    }
}

__global__ void meta_softmax(const float* __restrict__ mw, float* __restrict__ out) {
    int r = threadIdx.x;
    if (r < 4) {
        float a = mw[r * 3], b = mw[r * 3 + 1], c = mw[r * 3 + 2];
        float mx = fmaxf(a, fmaxf(b, c));
        float ea = __expf(a - mx), eb = __expf(b - mx), ec = __expf(c - mx);
        float inv = 1.f / (ea + eb + ec);
        out[r * 3] = ea * inv; out[r * 3 + 1] = eb * inv; out[r * 3 + 2] = ec * inv;
    }
}

// ---------------------------------------------------------------------------
extern "C" void kernel_launch(void* const* d_in, const int* in_sizes, int n_in,
                              void* d_out, int out_size, void* d_ws, size_t ws_size,
                              hipStream_t stream) {
    const float* x      = (const float*)d_in[0];
    const float* pos    = (const float*)d_in[1];
    const int*   eidx   = (const int*)  d_in[2];
    const float* W_in   = (const float*)d_in[3];
    const float* b_in   = (const float*)d_in[4];
    const float* Wq     = (const float*)d_in[5];
    const float* Wk     = (const float*)d_in[6];
    const float* Wv     = (const float*)d_in[7];
    const float* Wo     = (const float*)d_in[8];
    const float* ln1_g  = (const float*)d_in[9];
    const float* ln1_b  = (const float*)d_in[10];
    const float* ln2_g  = (const float*)d_in[11];
    const float* ln2_b  = (const float*)d_in[12];
    const float* ffn_w1 = (const float*)d_in[13];
    const float* ffn_b1 = (const float*)d_in[14];
    const float* ffn_w2 = (const float*)d_in[15];
    const float* ffn_b2 = (const float*)d_in[16];
    const float* meta_w = (const float*)d_in[17];
    const float* W_f1   = (const float*)d_in[18];
    const float* b_f1   = (const float*)d_in[19];
    const float* W_f2   = (const float*)d_in[20];
    const float* b_f2   = (const float*)d_in[21];

    float* ws = (float*)d_ws;
    const size_t NF = (size_t)NN * HID_;       // 6.4M floats
    float* h    = ws;
    float* q    = ws + NF;
    float* kbuf = ws + 2 * NF;
    float* vbuf = ws + 3 * NF;
    float* agg  = ws + 4 * NF;
    float* weff = ws + 5 * NF;

    const int nRT = NN / 16;                   // 3125
    const dim3 blk(256);
    const dim3 gGrid128((nRT + 7) / 8, HID_ / 16);
    const dim3 gGrid64 ((nRT + 7) / 8, OUT_ / 16);
    const int  lnGrid = (NN + 7) / 8;
    const unsigned ldsBytes = 128 * 516 * sizeof(float);  // 264192

    // h = concat(x, pos) @ W_in + b_in
    gemm_inproj<<<gGrid128, blk, 0, stream>>>(x, pos, W_in, b_in, h, nRT);

    for (int l = 0; l < 3; ++l) {
        const float* wq = Wq + (size_t)l * HID_ * HID_;
        const float* wk = Wk + (size_t)l * HID_ * HID_;
        const float* wv = Wv + (size_t)l * HID_ * HID_;
        const float* wo = Wo + (size_t)l * HID_ * HID_;

        gemm_wmma_k<128, false, false><<<gGrid128, blk, 0, stream>>>(h, wq, nullptr, q,    nRT, HID_);
        gemm_wmma_k<128, false, false><<<gGrid128, blk, 0, stream>>>(h, wk, nullptr, kbuf, nRT, HID_);
        gemm_wmma_k<128, false, false><<<gGrid128, blk, 0, stream>>>(h, wv, nullptr, vbuf, nRT, HID_);

        zero_f32<<<(unsigned)(NF / 256), 256, 0, stream>>>(agg, (long)NF);
        edge_attn<<<(EE + 7) / 8, 256, 0, stream>>>(eidx, eidx + EE, q, kbuf, vbuf, agg, EE);

        // o = agg @ Wo  (into q), then h = LN(h + o)
        gemm_wmma_k<128, false, false><<<gGrid128, blk, 0, stream>>>(agg, wo, nullptr, q, nRT, HID_);
        add_ln<<<lnGrid, blk, 0, stream>>>(h, q, ln1_g + l * HID_, ln1_b + l * HID_, NN);

        // f = relu(h@W1+b1)@W2+b2 (into kbuf), then h = LN(h + f)
        ffn_fused<<<(nRT + 7) / 8, blk, ldsBytes, stream>>>(
            h, ffn_w1 + (size_t)l * HID_ * 512, ffn_b1 + (size_t)l * 512,
            ffn_w2 + (size_t)l * 512 * HID_,    ffn_b2 + (size_t)l * HID_, kbuf, nRT);
        add_ln<<<lnGrid, blk, 0, stream>>>(h, kbuf, ln2_g + l * HID_, ln2_b + l * HID_, NN);
    }

    // Final layer: tile(h,4)@W_f1 == h@sum(W_f1 blocks)
    sum_wf1<<<64, 256, 0, stream>>>(W_f1, weff);
    gemm_wmma_k<128, true, true ><<<gGrid128, blk, 0, stream>>>(h, weff, b_f1, q, nRT, HID_);
    float* out = (float*)d_out;
    gemm_wmma_k<128, true, false><<<gGrid64,  blk, 0, stream>>>(q, W_f2, b_f2, out, nRT, OUT_);

    meta_softmax<<<1, 32, 0, stream>>>(meta_w, out + (size_t)NN * OUT_);
}